// _Score_Con_Global_58634893525775
// MI455X (gfx1250) — compile-verified
//
#include <hip/hip_runtime.h>

typedef float v2f __attribute__((ext_vector_type(2)));
typedef float v8f __attribute__((ext_vector_type(8)));

#define HDIM 512
#define TT   2048
#define BB   8
// virtual row-matrix: rows [0,16384) = enc (uses wv1), [16384,32768) = dec (uses wv2)
#define NROWS (2 * BB * TT)

// ---------------------------------------------------------------------------
// Kernel 1: wv[0..1023] = W(1024x512) @ v(512).  One wave32 per output row.
// ---------------------------------------------------------------------------
__global__ void wv_kernel(const float* __restrict__ W, const float* __restrict__ v,
                          float* __restrict__ wv) {
  const int gwave = (blockIdx.x * blockDim.x + threadIdx.x) >> 5;  // == row, 0..1023
  const int lane  = threadIdx.x & 31;
  const float* row = W + (size_t)gwave * HDIM;
  float acc = 0.f;
#pragma unroll 4
  for (int k = lane; k < HDIM; k += 32) acc += row[k] * v[k];
#pragma unroll
  for (int off = 16; off > 0; off >>= 1) acc += __shfl_down(acc, off, 32);
  if (lane == 0) wv[gwave] = acc;
}

// ---------------------------------------------------------------------------
// Kernel 2: row-dot-products via V_WMMA_F32_16X16X4_F32.
// Wave w owns 16 consecutive rows; K advances 4 per WMMA (128 WMMAs / wave).
// B = wv chunk broadcast across all 16 N columns -> every D column holds the
// full dot product; lanes 0 and 16 (N==0) extract rows 0-7 / 8-15.
// ---------------------------------------------------------------------------
__global__ void gemv_wmma_kernel(const float* __restrict__ enc,
                                 const float* __restrict__ dec,
                                 const float* __restrict__ wv,
                                 float* __restrict__ ed) {
  const int lane  = threadIdx.x & 31;
  const int gwave = blockIdx.x * (blockDim.x >> 5) + (threadIdx.x >> 5);
  const int row0  = gwave * 16;

  const float* src;
  const float* w;
  if (row0 < BB * TT) {                       // uniform per wave
    src = enc + (size_t)row0 * HDIM;
    w   = wv;                                 // wv1
  } else {
    src = dec + (size_t)(row0 - BB * TT) * HDIM;
    w   = wv + HDIM;                          // wv2
  }

  const int m  = lane & 15;                   // A-matrix M owned by this lane
  const int hi = lane >> 4;                   // 0: K={0,1}, 1: K={2,3}
  const float* rowp = src + (size_t)m * HDIM + 2 * hi;   // 8B aligned
  const float* wp   = w + 2 * hi;                        // 8B aligned

  v8f c = {};
#pragma unroll 4
  for (int kk = 0; kk < HDIM; kk += 4) {
    v2f a = *(const v2f*)(rowp + kk);         // A[m][kk+2*hi .. +1]
    v2f b = *(const v2f*)(wp + kk);           // B[k][n] = wv[k] for all n
    c = __builtin_amdgcn_wmma_f32_16x16x4_f32(
        /*neg_a=*/false, a, /*neg_b=*/false, b,
        /*c_mod=*/(short)0, c, /*reuse_a=*/false, /*reuse_b=*/false);
  }

  // D layout: VGPR p, lanes 0-15 -> M=p (N=lane), lanes 16-31 -> M=p+8.
  if (m == 0) {                               // lanes 0 and 16 hold N==0
    float* outp = ed + row0 + hi * 8;
#pragma unroll
    for (int p = 0; p < 8; ++p) outp[p] = c[p];
  }
}

// ---------------------------------------------------------------------------
// Kernel 3: out[b,i,j] = d[b,i] + e[b,j].  One block per (b,i) row.
// Pure streaming: float4 loads of the e row (L2-hot), b128 stores of output.
// ---------------------------------------------------------------------------
__global__ void bcast_add_kernel(const float* __restrict__ e,
                                 const float* __restrict__ d,
                                 float* __restrict__ out) {
  const int bi = blockIdx.x;          // b*2048 + i
  const int b  = bi >> 11;
  const float dv = d[bi];             // uniform -> scalar load
  const float4* erow = (const float4*)(e + (size_t)b * TT);
  float4* orow = (float4*)(out + (size_t)bi * TT);
  const int t = threadIdx.x;
#pragma unroll
  for (int r = 0; r < 2; ++r) {
    const int idx = t + r * 256;      // 256 thr * 2 * 4 floats = 2048 = row
    float4 ev = erow[idx];
    float4 ov = make_float4(ev.x + dv, ev.y + dv, ev.z + dv, ev.w + dv);
    orow[idx] = ov;
  }
}

// ---------------------------------------------------------------------------
extern "C" void kernel_launch(void* const* d_in, const int* in_sizes, int n_in,
                              void* d_out, int out_size, void* d_ws, size_t ws_size,
                              hipStream_t stream) {
  const float* enc = (const float*)d_in[0];   // (8, 2048, 512)
  const float* dec = (const float*)d_in[1];   // (8, 2048, 512)
  const float* W   = (const float*)d_in[2];   // (1024, 512)
  const float* v   = (const float*)d_in[3];   // (512)
  float* out = (float*)d_out;                 // (8, 2048, 2048)

  float* wv = (float*)d_ws;                   // 1024 floats: wv1 | wv2
  float* ed = wv + 2 * HDIM;                  // 32768 floats: e (16384) | d (16384)

  // K1: 1024 rows * 1 wave = 32768 threads
  wv_kernel<<<128, 256, 0, stream>>>(W, v, wv);
  // K2: 32768 rows / 16 per wave = 2048 waves = 65536 threads
  gemv_wmma_kernel<<<256, 256, 0, stream>>>(enc, dec, wv, ed);
  // K3: one block per (b,i) row
  bcast_add_kernel<<<BB * TT, 256, 0, stream>>>(ed, ed + BB * TT, out);
}